// Encoding_74371653698241
// MI455X (gfx1250) — compile-verified
//
#include <hip/hip_runtime.h>
#include <math.h>

// Problem constants (fixed by the reference harness)
#define BB   4
#define CCH  256
#define HW   4096
#define KKC  16
#define NCH  8
#define CHUNK (HW / NCH)   // 512

typedef __attribute__((ext_vector_type(2))) float v2f;
typedef __attribute__((ext_vector_type(8))) float v8f;

// ---- workspace layout (float offsets) ----
#define WS_CONV   0                               // b*K*hw conv / CW buffer
#define WS_BNA    (BB*KKC*HW)                     // 16: gamma*rsqrt(var+eps)
#define WS_BNSH   (WS_BNA + KKC)                  // 16: beta - mean*a
#define WS_SX     (WS_BNSH + KKC)                 // b*C
#define WS_SXX    (WS_SX + BB*CCH)                // b*C
#define WS_SCALE  (WS_SXX + BB*CCH)               // b*K  (row means of CW)
#define WS_SRP    (WS_SCALE + BB*KKC)             // b*K*16 per-row per-kk partials
#define WS_SRRP   (WS_SRP + BB*KKC*KKC)           // b*K*16
#define WS_SR     (WS_SRRP + BB*KKC*KKC)          // b*K
#define WS_SRR    (WS_SR + BB*KKC)                // b*K
#define WS_CROSSP (WS_SRR + BB*KKC)               // NCH*b*K*C partial Cross

__device__ inline v8f wmma4(v2f a, v2f b, v8f c) {
  return __builtin_amdgcn_wmma_f32_16x16x4_f32(false, a, false, b,
                                               (short)0, c, false, false);
}

// 1) conv[b,k,p] = sum_c W[k,c] * x[b,c,p]  -- f32 WMMA GEMM, one 16x16 tile/wave
__global__ __launch_bounds__(32)
void k_conv_wmma(const float* __restrict__ x, const float* __restrict__ w,
                 float* __restrict__ ws) {
  const int lane = threadIdx.x;
  const int l    = lane & 15;
  const int half = lane >> 4;
  const int tile  = blockIdx.x & 255;     // 256 position-tiles per batch
  const int batch = blockIdx.x >> 8;
  const int p0 = tile * 16;
  const float* Xb = x + (size_t)batch * CCH * HW;
  float* convb = ws + WS_CONV + (size_t)batch * KKC * HW;

  v8f acc = {0.f,0.f,0.f,0.f,0.f,0.f,0.f,0.f};
  for (int c0 = 0; c0 < CCH; c0 += 4) {
    const int ca = c0 + 2 * half;
    v2f a, b;
    a.x = w[l * CCH + ca];                       // A[m=l][k=ca]
    a.y = w[l * CCH + ca + 1];
    b.x = Xb[(size_t)ca * HW + p0 + l];          // B[k=ca][n=l]
    b.y = Xb[(size_t)(ca + 1) * HW + p0 + l];
    acc = wmma4(a, b, acc);
  }
#pragma unroll
  for (int j = 0; j < 8; ++j)                    // D: M=j+8*half -> k, N=l -> p
    convb[(size_t)(j + 8 * half) * HW + p0 + l] = acc[j];
}

// 2) BN batch stats over (b,hw) per k; fold gamma/beta into affine a, sh
__global__ __launch_bounds__(256)
void k_bn_stats(const float* __restrict__ gamma, const float* __restrict__ beta,
                float* __restrict__ ws) {
  const int k = blockIdx.x;
  const int tid = threadIdx.x;
  __shared__ float rs[256], rq[256];
  float s = 0.f, q = 0.f;
  for (int b = 0; b < BB; ++b) {
    const float* row = ws + WS_CONV + (size_t)(b * KKC + k) * HW;
    for (int i = tid; i < HW; i += 256) { float v = row[i]; s += v; q += v * v; }
  }
  rs[tid] = s; rq[tid] = q; __syncthreads();
  for (int off = 128; off > 0; off >>= 1) {
    if (tid < off) { rs[tid] += rs[tid + off]; rq[tid] += rq[tid + off]; }
    __syncthreads();
  }
  if (tid == 0) {
    const float n = (float)(BB * HW);
    float mean = rs[0] / n;
    float var  = rq[0] / n - mean * mean;
    float a = gamma[k] * rsqrtf(var + 1e-5f);
    ws[WS_BNA + k]  = a;
    ws[WS_BNSH + k] = beta[k] - mean * a;
  }
}

// 3) BN+ReLU in place; per-row per-(p&15) partial sums (flat-reshape buckets)
__global__ __launch_bounds__(256)
void k_bn_apply(float* __restrict__ ws) {
  const int bk  = blockIdx.x;          // b*16 + k
  const int k   = bk & 15;
  const int tid = threadIdx.x;
  float* row = ws + WS_CONV + (size_t)bk * HW;
  const float a  = ws[WS_BNA + k];
  const float sh = ws[WS_BNSH + k];
  float sr = 0.f, srr = 0.f;           // this thread's bucket is kk = tid & 15
  for (int i = tid; i < HW; i += 256) {
    float v = fmaxf(fmaf(row[i], a, sh), 0.f);
    row[i] = v;
    sr += v; srr += v * v;
  }
  __shared__ float ra[256], rb[256];
  ra[tid] = sr; rb[tid] = srr; __syncthreads();
  // strides >=16 keep (tid & 15) buckets separate
  for (int off = 128; off >= 16; off >>= 1) {
    if (tid < off) { ra[tid] += ra[tid + off]; rb[tid] += rb[tid + off]; }
    __syncthreads();
  }
  if (tid < 16) {
    ws[WS_SRP  + bk * KKC + tid] = ra[tid];
    ws[WS_SRRP + bk * KKC + tid] = rb[tid];
  }
}

// 4) scale[b,k] (row mean), SR[b,kk], SRR[b,kk] from partials
__global__ void k_bn_reduce(float* __restrict__ ws) {
  const int t = threadIdx.x;           // b*16 + k
  if (t < BB * KKC) {
    const int b = t >> 4, k = t & 15;
    float sc = 0.f, sr = 0.f, srr = 0.f;
    for (int j = 0; j < KKC; ++j) {
      sc  += ws[WS_SRP + t * KKC + j];                 // sum of row t's buckets
      sr  += ws[WS_SRP  + (b * KKC + j) * KKC + k];    // bucket k over all rows
      srr += ws[WS_SRRP + (b * KKC + j) * KKC + k];
    }
    ws[WS_SCALE + t] = sc * (1.f / (float)HW);
    ws[WS_SR + t]  = sr;
    ws[WS_SRR + t] = srr;
  }
}

// 5) Sx[b,c], Sxx[b,c]
__global__ __launch_bounds__(256)
void k_xstats(const float* __restrict__ x, float* __restrict__ ws) {
  const int bc = blockIdx.x;
  const int tid = threadIdx.x;
  const float* row = x + (size_t)bc * HW;
  float s = 0.f, q = 0.f;
  for (int i = tid; i < HW; i += 256) { float v = row[i]; s += v; q += v * v; }
  __shared__ float rs[256], rq[256];
  rs[tid] = s; rq[tid] = q; __syncthreads();
  for (int off = 128; off > 0; off >>= 1) {
    if (tid < off) { rs[tid] += rs[tid + off]; rq[tid] += rq[tid + off]; }
    __syncthreads();
  }
  if (tid == 0) { ws[WS_SX + bc] = rs[0]; ws[WS_SXX + bc] = rq[0]; }
}

// 6) Cross[b,k,c] = sum_p x[b,c,p] * CWflat[b][p*16+k]  -- f32 WMMA GEMM,
//    hw split into NCH chunks, per-chunk partials (deterministic, no atomics)
__global__ __launch_bounds__(32)
void k_cross_wmma(const float* __restrict__ x, float* __restrict__ ws) {
  const int lane = threadIdx.x;
  const int l    = lane & 15;
  const int half = lane >> 4;
  int id = blockIdx.x;
  const int ch = id & (NCH - 1); id >>= 3;
  const int ct = id & 15;        id >>= 4;
  const int batch = id;
  const int c0 = ct * 16;
  const int pbeg = ch * CHUNK;
  const float* Xb  = x + (size_t)batch * CCH * HW;
  const float* CWb = ws + WS_CONV + (size_t)batch * KKC * HW;  // flat (hw,K) view

  v8f acc = {0.f,0.f,0.f,0.f,0.f,0.f,0.f,0.f};
  for (int p0 = pbeg; p0 < pbeg + CHUNK; p0 += 4) {
    const int pa = p0 + 2 * half;
    v2f a, b;
    a.x = Xb[(size_t)(c0 + l) * HW + pa];        // A[m=l][k=pa-p0]
    a.y = Xb[(size_t)(c0 + l) * HW + pa + 1];
    b.x = CWb[(size_t)pa * KKC + l];             // B[k][n=l] = Rmat[p][k-code]
    b.y = CWb[(size_t)(pa + 1) * KKC + l];
    acc = wmma4(a, b, acc);
  }
  float* cp = ws + WS_CROSSP + (size_t)((ch * BB + batch) * KKC + l) * CCH;
#pragma unroll
  for (int j = 0; j < 8; ++j)                    // D: M -> c, N=l -> k
    cp[c0 + j + 8 * half] = acc[j];
}

// 7) scaled_l2 -> softmax over K -> encoded_feat (b,K,C)
__global__ __launch_bounds__(256)
void k_finalize(const float* __restrict__ ws, float* __restrict__ out) {
  const int gid = blockIdx.x * blockDim.x + threadIdx.x;   // b*C + c
  if (gid >= BB * CCH) return;
  const int b = gid >> 8;
  const int c = gid & 255;
  const float sx  = ws[WS_SX + gid];
  const float sxx = ws[WS_SXX + gid];
  float v[KKC], d[KKC];
  float mx = -3.4e38f;
  for (int k = 0; k < KKC; ++k) {
    float cr = 0.f;
    for (int ch = 0; ch < NCH; ++ch)
      cr += ws[WS_CROSSP + (size_t)((ch * BB + b) * KKC + k) * CCH + c];
    const int bk = b * KKC + k;
    float l2 = sxx - 2.f * cr + ws[WS_SRR + bk];
    float sv = ws[WS_SCALE + bk] * l2;
    v[k] = sv; mx = fmaxf(mx, sv);
    d[k] = sx - ws[WS_SR + bk];
  }
  float den = 0.f;
  for (int k = 0; k < KKC; ++k) { float e = expf(v[k] - mx); v[k] = e; den += e; }
  const float inv = 1.f / den;
  for (int k = 0; k < KKC; ++k)
    out[(size_t)(b * KKC + k) * CCH + c] = v[k] * inv * d[k];
}

extern "C" void kernel_launch(void* const* d_in, const int* in_sizes, int n_in,
                              void* d_out, int out_size, void* d_ws, size_t ws_size,
                              hipStream_t stream) {
  (void)in_sizes; (void)n_in; (void)out_size; (void)ws_size;
  const float* x     = (const float*)d_in[0];
  const float* w     = (const float*)d_in[1];
  const float* gamma = (const float*)d_in[2];
  const float* beta  = (const float*)d_in[3];
  float* ws  = (float*)d_ws;
  float* out = (float*)d_out;

  k_conv_wmma <<<BB * (HW / 16), 32, 0, stream>>>(x, w, ws);
  k_bn_stats  <<<KKC, 256, 0, stream>>>(gamma, beta, ws);
  k_bn_apply  <<<BB * KKC, 256, 0, stream>>>(ws);
  k_bn_reduce <<<1, 64, 0, stream>>>(ws);
  k_xstats    <<<BB * CCH, 256, 0, stream>>>(x, ws);
  k_cross_wmma<<<NCH * BB * (CCH / 16), 32, 0, stream>>>(x, ws);
  k_finalize  <<<(BB * CCH + 255) / 256, 256, 0, stream>>>(ws, out);
}